// ScalarSGC_57947698758291
// MI455X (gfx1250) — compile-verified
//
#include <hip/hip_runtime.h>

typedef __attribute__((ext_vector_type(2))) float v2f;
typedef __attribute__((ext_vector_type(8))) float v8f;

constexpr int NFEAT = 512;
constexpr int NHID  = 256;
constexpr int NOUT  = 64;

// One wave computes a 16(row) x 64(col) strip of C = A[nrows,K] @ W[K,NC] + bias.
// Uses V_WMMA_F32_16X16X4_F32 (fp32 in / fp32 acc), 4 accumulators share one A
// fragment per k-step. Wave-uniform guard keeps EXEC all-ones for WMMA.
// NTA: non-temporal loads of A (set for GEMM2 where agg is on its last use).
// NTC: non-temporal store of C (set for GEMM2 where out is never re-read).
template <int K, int NC, bool NTA, bool NTC>
__global__ __launch_bounds__(128) void sgc_wmma_gemm_bias(
    const float* __restrict__ A, const float* __restrict__ W,
    const float* __restrict__ bias, float* __restrict__ C, int nrows) {
  const int lane = threadIdx.x & 31;
  const int wave = threadIdx.x >> 5;
  const int rowTile = blockIdx.y * 4 + wave;   // 16 rows per tile
  const int ntiles = nrows >> 4;
  if (rowTile >= ntiles) return;               // wave-uniform: EXEC stays full

  const int row0 = rowTile << 4;
  const int col0 = blockIdx.x << 6;            // 64-wide column strip
  const int m  = lane & 15;                    // M for A, N for B/C/D
  const int kh = lane >> 4;                    // lane-half selects K pair / row half

  v8f acc[4] = {};

  // A fragment: lane holds A[row0+m][kb + 2*kh + {0,1}]  (ISA 16x4 f32 layout)
  const float* arow = A + (size_t)(row0 + m) * K + 2 * kh;

  for (int kb = 0; kb < K; kb += 4) {
    v2f a;
    if (NTA) {
      a.x = __builtin_nontemporal_load(&arow[kb]);
      a.y = __builtin_nontemporal_load(&arow[kb + 1]);
    } else {
      a.x = arow[kb];
      a.y = arow[kb + 1];
    }
    // B fragment: lane holds W[kb + 2*kh + v][col0 + t*16 + m]
    const float* wp = W + (size_t)(kb + 2 * kh) * NC + col0 + m;
#pragma unroll
    for (int t = 0; t < 4; ++t) {
      v2f b;
      b.x = wp[t * 16];
      b.y = wp[t * 16 + NC];
      acc[t] = __builtin_amdgcn_wmma_f32_16x16x4_f32(
          false, a, false, b, (short)0, acc[t], false, false);
    }
  }

  // C/D layout: VGPR r -> row (r + 8*kh), col = lane&15 within each 16-tile.
#pragma unroll
  for (int t = 0; t < 4; ++t) {
    const int col = col0 + t * 16 + m;
    const float bv = bias[col];
#pragma unroll
    for (int r = 0; r < 8; ++r) {
      const int row = row0 + r + 8 * kh;
      float outv = acc[t][r] + bv;
      if (NTC)
        __builtin_nontemporal_store(outv, &C[(size_t)row * NC + col]);
      else
        C[(size_t)row * NC + col] = outv;
    }
  }
}

// Edge-parallel SpMM scatter: agg[r,:] += val * h[c,:].
// One wave per edge; 256 feats / 32 lanes = 8 coalesced f32 atomics per lane.
// h and agg (102 MB each) both live in the 192 MB L2, so the gather + atomic
// scatter traffic stays on-chip; atomics are no-return (STOREcnt path).
// Edge arrays are single-pass -> non-temporal loads keep L2 for h/agg.
__global__ __launch_bounds__(256) void sgc_spmm_scatter(
    const float* __restrict__ h, const float* __restrict__ val,
    const int* __restrict__ row, const int* __restrict__ col,
    float* __restrict__ agg, int ne) {
  const int lane = threadIdx.x & 31;
  const int gwave = (blockIdx.x * blockDim.x + threadIdx.x) >> 5;
  const int nwaves = (gridDim.x * blockDim.x) >> 5;

  for (int e = gwave; e < ne; e += nwaves) {
    const int r = __builtin_nontemporal_load(&row[e]);
    const int c = __builtin_nontemporal_load(&col[e]);
    const float v = __builtin_nontemporal_load(&val[e]);
    const float* hp = h + (size_t)c * NHID;
    float* ap = agg + (size_t)r * NHID;
#pragma unroll
    for (int i = 0; i < NHID / 32; ++i) {
      atomicAdd(&ap[i * 32 + lane], v * hp[i * 32 + lane]);
    }
  }
}

extern "C" void kernel_launch(void* const* d_in, const int* in_sizes, int n_in,
                              void* d_out, int out_size, void* d_ws,
                              size_t ws_size, hipStream_t stream) {
  const float* x       = (const float*)d_in[0];
  const float* adj_val = (const float*)d_in[1];
  const float* W1      = (const float*)d_in[2];
  const float* b1      = (const float*)d_in[3];
  const float* W2      = (const float*)d_in[4];
  const float* b2      = (const float*)d_in[5];
  const int*   adj_row = (const int*)d_in[6];
  const int*   adj_col = (const int*)d_in[7];
  float* out = (float*)d_out;

  const int N  = in_sizes[0] / NFEAT;  // 100000
  const int NE = in_sizes[1];          // 3200000

  float* h   = (float*)d_ws;                  // [N, NHID]
  float* agg = h + (size_t)N * NHID;          // [N, NHID]

  const dim3 blk(128);
  const int rowBlocks = ((N >> 4) + 3) / 4;

  // h = x @ W1 + b1   (x streamed; h stays L2-resident for the SpMM gather)
  sgc_wmma_gemm_bias<NFEAT, NHID, false, false>
      <<<dim3(NHID / 64, rowBlocks), blk, 0, stream>>>(x, W1, b1, h, N);

  // agg = 0
  hipMemsetAsync(agg, 0, (size_t)N * NHID * sizeof(float), stream);

  // agg[r] += val * h[c]
  sgc_spmm_scatter<<<4096, 256, 0, stream>>>(h, adj_val, adj_row, adj_col, agg, NE);

  // out = agg @ W2 + b2  (agg last-use -> NT loads; out never re-read -> NT store)
  sgc_wmma_gemm_bias<NHID, NOUT, true, true>
      <<<dim3(NOUT / 64, rowBlocks), blk, 0, stream>>>(agg, W2, b2, out, N);
}